// Feature_16621523436259
// MI455X (gfx1250) — compile-verified
//
#include <hip/hip_runtime.h>
#include <cmath>

// ============================================================================
// ACSF (Behler-Parrinello symmetry functions) for MI455X / gfx1250, wave32.
//
// Design (reasoned from MI455X specs, compile-only):
//  * Problem: B*N = 4096 atoms, M = 32 neighbors, P = 496 pairs/atom.
//    Memory traffic ~2.8 MB total -> ~0.12 us at 23.3 TB/s. Transcendentals
//    dominate: ~37M exp/log after folding 2^(1-z)*base^z*exp(-eta*R2) into a
//    single exp(z*ln(base) - eta*R2 + (1-z)*ln2). => transcendental-bound.
//  * One wave32 per atom (wave32 is the native CDNA5 width). Neighbor
//    (r, vx, vy, vz) table staged in per-wave-private LDS; pairs enumerated
//    as cyclic offsets d=1..15 (32 pairs each) + d=16 (16 pairs, half-masked)
//    = 496 exactly, j=min/k=max to match np.triu order (needed because the
//    reference's f_rjk = fc(rik) bug breaks j<->k symmetry).
//  * Lane-axis reductions (sum of 32x{16,32} partials over the 32 lanes) are
//    done with V_WMMA_F32_16X16X4_F32 at full f32 precision: A = partials in
//    the documented 16x4 f32 A layout (VGPR0=K0/K2, VGPR1=K1/K3 per lane
//    half), B = all-ones (layout-invariant), 8 chained K=4 steps sum 32 lanes.
//    Partials are stored transposed in LDS so each A fragment is one
//    ds_load_b64 (compiler fuses pairs into ds_load_2addr_b64). CDNA5
//    split-counter fences (s_wait_dscnt) order the wave-private LDS staging.
//  * Output: D rows are contiguous in the v8f accumulator, so lanes 0 and 16
//    (one exec region, (lane&15)==0) emit 16-byte global_store_b128 writes --
//    12 wide stores per atom instead of 48 scalarized b32 stores.
// ============================================================================

typedef __attribute__((ext_vector_type(2))) float v2f;
typedef __attribute__((ext_vector_type(8))) float v8f;

constexpr int S_  = 8;
constexpr int F1_ = 32;
constexpr int F3_ = 16;
constexpr int B_  = 8;
constexpr int N_  = 512;
constexpr int M_  = 32;
constexpr int NATOM = B_ * N_;           // 4096 atoms == 4096 waves
constexpr int WAVES_PER_BLOCK = 8;       // 256 threads
constexpr int LDS_PER_WAVE = 128 + 32 * 32;  // floats: vec table + partials

constexpr float PI_F  = 3.14159265358979323846f;
constexpr float LN2_F = 0.69314718055994530942f;

#define LDS_FENCE() asm volatile("s_wait_dscnt 0" ::: "memory")

__device__ __forceinline__ float fcut(float r) {
    // smooth_cosine(r, 0, 6): t = clip(r/6, 0, 1); 0.5*(cos(pi t)+1)
    float t = fminf(fmaxf(r * (1.0f / 6.0f), 0.0f), 1.0f);
    return 0.5f * (__cosf(PI_F * t) + 1.0f);
}

// Sum 32 "m" columns for 16 feature rows stored transposed in LDS:
// partT[f*32 + m], f = 0..15, m = 0..31. Uses V_WMMA_F32_16X16X4_F32 with
// A[f][k] = partT[f][4*kc + k] and B = ones (all D columns equal the row sum).
// A-layout (ISA 7.12.2, 32-bit A 16x4): VGPR0 = {K0 lanes0-15, K2 lanes16-31},
// VGPR1 = {K1, K3}; M = lane & 15. One ds_load_b64 per chunk supplies (a0,a1).
// Result: lane 0 holds f=0..7 in d[0..7]; lane 16 holds f=8..15 in d[0..7].
__device__ __forceinline__ v8f wmma_colsum16(const float* partT, int lane) {
    v8f c = {0.f, 0.f, 0.f, 0.f, 0.f, 0.f, 0.f, 0.f};
    v2f b = {1.0f, 1.0f};                 // all-ones B, any layout works
    const int frow = lane & 15;
    const int moff = (lane & 16) ? 2 : 0; // lanes 16-31 supply K=2,3
    const float* rowp = partT + frow * 32 + moff;
#pragma unroll
    for (int kc = 0; kc < 8; ++kc) {
        v2f a = *(const v2f*)(rowp + 4 * kc);   // ds_load_b64
        c = __builtin_amdgcn_wmma_f32_16x16x4_f32(
            /*neg_a=*/false, a, /*neg_b=*/false, b,
            /*c_mod=*/(short)0, c, /*reuse_a=*/false, /*reuse_b=*/false);
    }
    return c;
}

__device__ __forceinline__ void store8(float* p, const v8f& d) {
    // Two 16B-aligned global_store_b128 writes of d[0..7].
    *(float4*)(p + 0) = make_float4(d[0], d[1], d[2], d[3]);
    *(float4*)(p + 4) = make_float4(d[4], d[5], d[6], d[7]);
}

__global__ __launch_bounds__(256, 1) void acsf_g1g3_kernel(
    const float* __restrict__ Rij,       // (B,N,M,4) = [r, vx, vy, vz]
    const int*   __restrict__ Zi,        // (B,N)
    const float* __restrict__ g1_eta,    // (S,F1)
    const float* __restrict__ g1_rs,     // (S,F1)
    const float* __restrict__ g3_eta,    // (S,F3)
    const float* __restrict__ g3_lambd,  // (F3)
    const float* __restrict__ g3_theta,  // (F3)
    float*       __restrict__ out)       // (B,N,F1+F3)
{
    __shared__ __align__(16) float smem[WAVES_PER_BLOCK * LDS_PER_WAVE];

    const int lane = threadIdx.x & 31;
    const int wave = threadIdx.x >> 5;
    const int atom = blockIdx.x * WAVES_PER_BLOCK + wave;  // grid sized exactly

    float*  wbase = smem + wave * LDS_PER_WAVE;
    float4* vt    = (float4*)wbase;       // 32 x (r, vx, vy, vz)
    float*  partT = wbase + 128;          // transposed partials [f][m]

    // --- stage this atom's neighbor list: lane m holds neighbor m ---
    const float4* rp = (const float4*)Rij + (size_t)atom * M_;
    __builtin_prefetch(rp, 0, 1);         // global_prefetch_b8
    float4 me = rp[lane];
    vt[lane] = me;
    LDS_FENCE();

    int sp = Zi[atom];
    sp = __builtin_amdgcn_readfirstlane(sp);   // wave-uniform -> scalar loads

    // ================= G1 (radial, F1 = 32 features) =================
    const float r  = me.x;
    const float fr = fcut(r);
    const float* e1 = g1_eta + sp * F1_;
    const float* r1 = g1_rs  + sp * F1_;
#pragma unroll
    for (int f = 0; f < F1_; ++f) {
        float d = r - r1[f];
        partT[f * 32 + lane] = __expf(-e1[f] * d * d) * fr;  // partT[f][m]
    }
    LDS_FENCE();

    v8f dA = wmma_colsum16(partT,            lane);  // features 0..15
    v8f dB = wmma_colsum16(partT + 16 * 32,  lane);  // features 16..31

    // Lane 0 holds f=0..7 (dA) / f=16..23 (dB); lane 16 holds f=8..15 / 24..31.
    float* ob = out + (size_t)atom * (F1_ + F3_);
    if ((lane & 15) == 0) {               // lanes 0 and 16, one exec region
        const int h = (lane >> 4) * 8;    // 0 or 8
        store8(ob + h,      dA);
        store8(ob + 16 + h, dB);
    }

    // ================= G3 (angular, F3 = 16 features) =================
    const float* e3 = g3_eta + sp * F3_;
    float zt[F3_], et[F3_], cf[F3_];
    bool  lp[F3_];
#pragma unroll
    for (int f = 0; f < F3_; ++f) {
        zt[f] = g3_theta[f];
        et[f] = e3[f];
        cf[f] = (1.0f - zt[f]) * LN2_F;      // ln(2^(1-zeta))
        lp[f] = g3_lambd[f] > 0.0f;
    }
    float acc[F3_];
#pragma unroll
    for (int f = 0; f < F3_; ++f) acc[f] = 0.0f;

    // Pair enumeration: d=1..15 -> 32 distinct unordered pairs each (480);
    // d=16 -> 16 pairs from lanes 0..15 (lanes 16..31 masked). Total 496.
    for (int d = 1; d <= 16; ++d) {
        int o = (lane + d) & 31;
        int j = lane < o ? lane : o;       // triu order: j < k
        int k = lane < o ? o : lane;
        float4 vj = vt[j];
        float4 vk = vt[k];
        float rij = vj.x, rik = vk.x;      // stored norms == recomputed norms
        float dx = vj.y - vk.y, dy = vj.z - vk.z, dz = vj.w - vk.w;
        float rjk2  = dx * dx + dy * dy + dz * dz;
        float dotjk = vj.y * vk.y + vj.z * vk.z + vj.w * vk.w;
        float cosine = dotjk / (rij * rik);
        float R2 = rij * rij + rik * rik + rjk2;
        float fik = fcut(rik);
        float F = fcut(rij) * fik * fik;   // reference bug: f_rjk = fc(rik)
        if (d == 16 && lane >= 16) F = 0.0f;   // mask duplicate half
        float lbp = __logf(fmaxf(1.0f + cosine, 0.0f));  // log(base | lam=+1)
        float lbm = __logf(fmaxf(1.0f - cosine, 0.0f));  // log(base | lam=-1)
#pragma unroll
        for (int f = 0; f < F3_; ++f) {
            float sb = lp[f] ? lbp : lbm;
            // 2^(1-z) * base^z * exp(-eta*R2) == exp(z*ln(base) - eta*R2 + cf)
            acc[f] += __expf(zt[f] * sb - et[f] * R2 + cf[f]) * F;
        }
    }

#pragma unroll
    for (int f = 0; f < F3_; ++f) partT[f * 32 + lane] = acc[f];
    LDS_FENCE();

    v8f d3 = wmma_colsum16(partT, lane);
    if ((lane & 15) == 0) {
        const int h = (lane >> 4) * 8;
        store8(ob + 32 + h, d3);
    }
}

extern "C" void kernel_launch(void* const* d_in, const int* in_sizes, int n_in,
                              void* d_out, int out_size, void* d_ws, size_t ws_size,
                              hipStream_t stream) {
    const float* Rij      = (const float*)d_in[0];
    const int*   Zi       = (const int*)  d_in[1];
    const float* g1_eta   = (const float*)d_in[2];
    const float* g1_rs    = (const float*)d_in[3];
    const float* g3_eta   = (const float*)d_in[4];
    const float* g3_lambd = (const float*)d_in[5];
    const float* g3_theta = (const float*)d_in[6];
    // d_in[7] (g3_rs) is unused by the reference forward pass.
    float* out = (float*)d_out;

    dim3 grid(NATOM / WAVES_PER_BLOCK);   // 512 blocks x 256 threads = 4096 waves
    dim3 block(WAVES_PER_BLOCK * 32);
    acsf_g1g3_kernel<<<grid, block, 0, stream>>>(
        Rij, Zi, g1_eta, g1_rs, g3_eta, g3_lambd, g3_theta, out);
}